// TopkSAE_48498770706813
// MI455X (gfx1250) — compile-verified
//
#include <hip/hip_runtime.h>

typedef float v2f __attribute__((ext_vector_type(2)));
typedef float v8f __attribute__((ext_vector_type(8)));
typedef int   v4i __attribute__((ext_vector_type(4)));

#define BM 64
#define BN 256
#define BK 16
#define APAD 20   // padded LDS row length (floats) for BK=16
#define BPAD 20

#define AS1 __attribute__((address_space(1)))
#define AS3 __attribute__((address_space(3)))

#if defined(__has_builtin)
#if __has_builtin(__builtin_amdgcn_global_load_async_to_lds_b128)
#define HAVE_ASYNC_LDS 1
#endif
#if __has_builtin(__builtin_amdgcn_s_wait_asynccnt)
#define HAVE_WAIT_ASYNC 1
#endif
#endif
#ifndef HAVE_ASYNC_LDS
#define HAVE_ASYNC_LDS 0
#endif
#ifndef HAVE_WAIT_ASYNC
#define HAVE_WAIT_ASYNC 0
#endif

__device__ __forceinline__ void wait_async0()
{
#if HAVE_ASYNC_LDS
#if HAVE_WAIT_ASYNC
    __builtin_amdgcn_s_wait_asynccnt(0);
#else
    asm volatile("s_wait_asynccnt 0x0" ::: "memory");
#endif
#endif
}

// ---------------------------------------------------------------------------
// GEMM: C[M,N] = (A[M,K] - a_sub[K]) * B[N,K]^T + bias[N]
// M%64==0, N%256==0, K%16==0 assumed.
// 256 threads = 8 wave32 waves arranged 2(M) x 4(N); wave tile = 32x64
// = 2x4 tiles of v_wmma_f32_16x16x4_f32.  B tile streamed with
// global_load_async_to_lds_b128 into a double-buffered LDS pipeline.
// ---------------------------------------------------------------------------
template <bool SUB>
__global__ __launch_bounds__(256)
void sae_gemm_kernel(const float* __restrict__ A,
                     const float* __restrict__ B,
                     const float* __restrict__ a_sub,
                     const float* __restrict__ bias,
                     float* __restrict__ C,
                     int M, int N, int K)
{
    __shared__ __align__(16) float Alds[2][BM * APAD];
    __shared__ __align__(16) float Blds[2][BN * BPAD];

    const int tid  = threadIdx.x;
    const int wave = tid >> 5;
    const int lane = tid & 31;
    const int wm   = wave >> 2;          // 0..1  (M direction)
    const int wn   = wave & 3;           // 0..3  (N direction)
    const int lhalf = lane >> 4;         // K-pair select (A/B), M-half (C)
    const int lidx  = lane & 15;         // m (A frag) or n (B frag / C col)

    const int block_m = blockIdx.y * BM;
    const int block_n = blockIdx.x * BN;

    // global staging map: 4 consecutive floats per thread per 64-row slab
    const int grow = tid >> 2;           // 0..63
    const int gcol = (tid & 3) << 2;     // 0,4,8,12

    const float* Arow = A + (size_t)(block_m + grow) * K;

    const v8f vzero = {0.f, 0.f, 0.f, 0.f, 0.f, 0.f, 0.f, 0.f};
    v8f acc[2][4];
#pragma unroll
    for (int i = 0; i < 2; ++i)
#pragma unroll
        for (int j = 0; j < 4; ++j) acc[i][j] = vzero;

    // stage one K-chunk into LDS buffer `buf`
    auto stage_chunk = [&](int buf, int k0) {
        // ---- B tile: 4x 16B per thread, async straight to LDS ----------
#pragma unroll
        for (int i = 0; i < 4; ++i) {
            const float* src = B + (size_t)(block_n + grow + 64 * i) * K + k0 + gcol;
            float*       dst = &Blds[buf][(grow + 64 * i) * BPAD + gcol];
#if HAVE_ASYNC_LDS
            __builtin_amdgcn_global_load_async_to_lds_b128(
                (AS1 v4i*)(void*)src, (AS3 v4i*)(void*)dst, 0, 0);
#else
            *(float4*)dst = *(const float4*)src;
#endif
        }
        // ---- A tile: via VGPRs so the (x - pre_bias) fusion is free ----
        float4 av = *(const float4*)(Arow + k0 + gcol);
        if (SUB) {
            const float4 pb = *(const float4*)(a_sub + k0 + gcol);
            av.x -= pb.x; av.y -= pb.y; av.z -= pb.z; av.w -= pb.w;
        }
        *(float4*)&Alds[buf][grow * APAD + gcol] = av;
    };

    const int nchunks = K / BK;

    stage_chunk(0, 0);
    wait_async0();
    __syncthreads();

    for (int c = 0; c < nchunks; ++c) {
        const int buf = c & 1;

        // prefetch next chunk into the other buffer (nobody reads it now)
        if (c + 1 < nchunks)
            stage_chunk(buf ^ 1, (c + 1) * BK);

        const float* Ab = &Alds[buf][0];
        const float* Bb = &Blds[buf][0];

        // ---- compute: 4 k-steps of K=4, 8 WMMA tiles each --------------
#pragma unroll
        for (int kk = 0; kk < 4; ++kk) {
            const int kb = kk * 4 + 2 * lhalf;   // this lane's K pair
            v2f afrag[2], bfrag[4];
#pragma unroll
            for (int i = 0; i < 2; ++i) {
                const int m = wm * 32 + i * 16 + lidx;
                afrag[i] = *(const v2f*)&Ab[m * APAD + kb];
            }
#pragma unroll
            for (int j = 0; j < 4; ++j) {
                const int n = wn * 64 + j * 16 + lidx;
                bfrag[j] = *(const v2f*)&Bb[n * BPAD + kb];
            }
#pragma unroll
            for (int i = 0; i < 2; ++i)
#pragma unroll
                for (int j = 0; j < 4; ++j)
                    acc[i][j] = __builtin_amdgcn_wmma_f32_16x16x4_f32(
                        false, afrag[i], false, bfrag[j],
                        (short)0, acc[i][j], false, false);
        }

        if (c + 1 < nchunks) {
            wait_async0();     // own async writes to next buffer complete
            __syncthreads();   // all waves: writes visible, reads finished
        }
    }

    // ---- epilogue: + bias[n], store -------------------------------------
    float bn[4];
#pragma unroll
    for (int j = 0; j < 4; ++j) {
        const int ng = block_n + wn * 64 + j * 16 + lidx;
        bn[j] = bias[ng];
    }
#pragma unroll
    for (int i = 0; i < 2; ++i) {
#pragma unroll
        for (int j = 0; j < 4; ++j) {
            const int ng = block_n + wn * 64 + j * 16 + lidx;
#pragma unroll
            for (int v = 0; v < 8; ++v) {
                const int mg = block_m + wm * 32 + i * 16 + v + 8 * lhalf;
                C[(size_t)mg * N + ng] = acc[i][j][v] + bn[j];
            }
        }
    }
}

// ---------------------------------------------------------------------------
// In-place TopK per row: keep the k largest of L values, zero the rest.
// One workgroup per token; the 128KB row lives in LDS (CDNA5: 320KB/WGP).
// 4-pass byte-radix select on monotonic keys, then threshold write-back.
// ---------------------------------------------------------------------------
__device__ __forceinline__ unsigned int f32_key(float f)
{
    unsigned int u = __float_as_uint(f);
    return (u & 0x80000000u) ? ~u : (u | 0x80000000u);   // order-preserving
}

__global__ __launch_bounds__(256)
void sae_topk_kernel(float* __restrict__ latents, const int* __restrict__ kptr, int L)
{
    extern __shared__ float row[];                   // L floats (dynamic LDS)
    __shared__ unsigned int hist[256];
    __shared__ unsigned int s_sel, s_want, s_tie;

    const int tid = threadIdx.x;
    float* grow = latents + (size_t)blockIdx.x * L;

    for (int i = tid; i < L; i += 256) row[i] = grow[i];
    __syncthreads();

    const unsigned int k = (unsigned int)(*kptr);
    unsigned int prefix = 0;
    unsigned int want   = k;

    for (int b = 3; b >= 0; --b) {
        hist[tid] = 0;
        __syncthreads();
        for (int i = tid; i < L; i += 256) {
            const unsigned int key = f32_key(row[i]);
            const unsigned int hi =
                (unsigned int)(((unsigned long long)key) >> (8 * (b + 1)));
            if (hi == prefix)
                atomicAdd(&hist[(key >> (8 * b)) & 255u], 1u);
        }
        __syncthreads();
        if (tid == 0) {
            unsigned int c = 0, sel = 0, w = want;
            for (int v = 255; v >= 0; --v) {
                c += hist[v];
                if (c >= w) { sel = (unsigned int)v; w = w - (c - hist[v]); break; }
            }
            s_sel = sel; s_want = w;
        }
        __syncthreads();
        prefix = (prefix << 8) | s_sel;
        want   = s_want;
    }

    if (tid == 0) s_tie = 0;
    __syncthreads();

    const unsigned int T   = prefix;   // exact key of the k-th largest
    const unsigned int rem = want;     // how many ties at T to keep
    for (int i = tid; i < L; i += 256) {
        const float f = row[i];
        const unsigned int key = f32_key(f);
        float outv = 0.0f;
        if (key > T) {
            outv = f;
        } else if (key == T) {
            const unsigned int pos = atomicAdd(&s_tie, 1u);
            if (pos < rem) outv = f;
        }
        grow[i] = outv;
    }
}

// ---------------------------------------------------------------------------
extern "C" void kernel_launch(void* const* d_in, const int* in_sizes, int n_in,
                              void* d_out, int out_size, void* d_ws, size_t ws_size,
                              hipStream_t stream)
{
    const float* x           = (const float*)d_in[0];
    const float* pre_bias    = (const float*)d_in[1];
    const float* latent_bias = (const float*)d_in[2];
    const float* enc_w       = (const float*)d_in[3];   // [L, H]
    const float* dec_w       = (const float*)d_in[4];   // [H, L]
    const int*   kptr        = (const int*)d_in[5];

    const int H  = 4096;
    const int L  = 32768;
    const int NT = in_sizes[0] / H;                     // tokens = B*S = 4096

    float* latents = (float*)d_out;                     // [NT, L]
    float* xhat    = (float*)d_out + (size_t)NT * L;    // [NT, H]

    dim3 blk(256);

    // 1) encode: latents(dense) = (x - pre_bias) @ enc_w^T + latent_bias
    sae_gemm_kernel<true><<<dim3(L / BN, NT / BM), blk, 0, stream>>>(
        x, enc_w, pre_bias, latent_bias, latents, NT, L, H);

    // 2) top-k sparsify each row of latents in place
    sae_topk_kernel<<<dim3(NT), blk, (size_t)L * sizeof(float), stream>>>(
        latents, kptr, L);

    // 3) decode (dense GEMM keeps dec_w reads coalesced):
    //    xhat = latents @ dec_w^T + pre_bias
    sae_gemm_kernel<false><<<dim3(H / BN, NT / BM), blk, 0, stream>>>(
        latents, dec_w, nullptr, pre_bias, xhat, NT, H, L);
}